// RMAC_85263690760822
// MI455X (gfx1250) — compile-verified
//
#include <hip/hip_runtime.h>
#include <hip/hip_bf16.h>

// R-MAC pooling on x[64, 2048, 16, 16] expressed as a K=256 GEMM against a
// constant 14x256 region-weight matrix, using V_WMMA_F32_16X16X4_F32.
//   A (weight): lane (m+16s) holds W[region m][k0+2s .. +1]  (from LDS table)
//   B (data):   lane (n+16s) holds x[row n][k0+2s .. k0+2s+1] (float2 load)
//   D: vgpr r, lanes 0-15 -> region r, rows 0..15 ; lanes 16-31 -> region r+8
// Regions 14,15 carry zero weight -> their accumulators are exactly 0.
// The 64-step A-fragment sequence is identical for every wave, so it is
// built once per block into LDS (64 steps x 32 lanes x float2 = 16 KB) and
// the steady-state loop is: ds_load_b64 + global_load_b64 + v_wmma.

typedef float v2f __attribute__((ext_vector_type(2)));
typedef float v8f __attribute__((ext_vector_type(8)));

#define BATCH       64
#define CHANS       2048
#define REG_TOTAL   14
#define OUT_COLS    (REG_TOTAL * CHANS)       // 28672
#define ROWS        (BATCH * CHANS)           // 131072 (b,c) rows of 256 pixels
#define WAVES       (ROWS / 16)               // 8192
#define WAVES_PER_B (CHANS / 16)              // 128
#define KSTEPS      64                        // 256 / 4

__global__ void rmac_pool_wmma(const float* __restrict__ x,
                               float* __restrict__ out,
                               float* __restrict__ partials) {
    __shared__ v2f wtab[KSTEPS * 32];         // [step][lane] A fragments, 16 KB

    // ---- build the shared weight-fragment table (identical for all waves) ----
    for (int idx = threadIdx.x; idx < KSTEPS * 32; idx += blockDim.x) {
        const int lw = idx & 31;
        const int mm = lw & 15;               // region id
        const int sw = lw >> 4;               // K sub-slot
        const int p  = ((idx >> 5) << 2) + 2 * sw;   // pixel index (and p+1)
        int h0 = 0, w0 = 0, len = 0; float wt = 0.0f;
        if (mm == 0)       { len = 16; wt = 1.0f / 256.0f; }
        else if (mm <= 4)  { int t = mm - 1; h0 = (t >> 1) * 4; w0 = (t & 1) * 4; len = 10; wt = 1.0f / 100.0f; }
        else if (mm <= 13) { int t = mm - 5; h0 = (t / 3) * 3;  w0 = (t % 3) * 3; len = 8;  wt = 1.0f / 64.0f;  }
        const unsigned hm = ((1u << len) - 1u) << h0;
        const unsigned wm = ((1u << len) - 1u) << w0;
        const int h = p >> 4;
        const int w = p & 15;                 // p even -> w even -> w+1 same row
        const unsigned hb = (hm >> h) & 1u;
        v2f a;
        a.x = ((hb & (wm >> w)) & 1u)       ? wt : 0.0f;
        a.y = ((hb & (wm >> (w + 1))) & 1u) ? wt : 0.0f;
        wtab[idx] = a;
    }
    __syncthreads();

    const int lane     = threadIdx.x & 31;
    const int wid      = blockIdx.x * (blockDim.x >> 5) + (threadIdx.x >> 5); // 0..8191
    const int row_base = wid << 4;            // 16 (b,c) rows per wave
    const int n        = lane & 15;           // row-in-group (B) / channel lane (store)
    const int s        = lane >> 4;           // K sub-slot

    const float* rowp = x + (size_t)(row_base + n) * 256 + 2 * s;
    const v2f*   wrow = wtab + lane;

    v8f acc = {0.f, 0.f, 0.f, 0.f, 0.f, 0.f, 0.f, 0.f};

#pragma unroll
    for (int step = 0; step < KSTEPS; ++step) {
        const v2f a = wrow[step * 32];                    // ds_load_b64, imm offset
        const v2f b = *(const v2f*)(rowp + step * 4);     // global_load_b64, imm offset
        acc = __builtin_amdgcn_wmma_f32_16x16x4_f32(
            /*neg_a=*/false, a, /*neg_b=*/false, b,
            /*c_mod=*/(short)0, acc, /*reuse_a=*/false, /*reuse_b=*/false);
    }

    // ---- store D + accumulate squared norm ----
    const int b_idx  = row_base >> 11;        // /CHANS (waves never straddle a batch)
    const int c_base = row_base & (CHANS - 1);
    float* obase = out + (size_t)b_idx * OUT_COLS + c_base + n;

    float ss = 0.0f;
#pragma unroll
    for (int r = 0; r < 6; ++r) {             // g = r + 8s <= 13 for all lanes
        const float v = acc[r];
        ss += v * v;
        obase[(size_t)(r + 8 * s) * CHANS] = v;
    }
    {   // r = 6,7: valid only for lanes 0-15 (g = 6,7); g = 14,15 are zero rows
        const float v6 = acc[6], v7 = acc[7];
        ss += v6 * v6 + v7 * v7;              // exact zeros on s==1 lanes
        if (s == 0) {
            obase[(size_t)6 * CHANS] = v6;
            obase[(size_t)7 * CHANS] = v7;
        }
    }

    // wave32 reduction of squared sums (all lanes share the same b)
#pragma unroll
    for (int off = 16; off > 0; off >>= 1)
        ss += __shfl_xor(ss, off, 32);
    if (lane == 0)
        partials[wid] = ss;
}

// Deterministic tree reduction: 128 wave-partials per batch row -> norms[b]
__global__ void rmac_norm_reduce(const float* __restrict__ partials,
                                 float* __restrict__ norms) {
    __shared__ float sdata[WAVES_PER_B];
    const int b = blockIdx.x;
    const int t = threadIdx.x;
    sdata[t] = partials[b * WAVES_PER_B + t];
    __syncthreads();
#pragma unroll
    for (int off = WAVES_PER_B / 2; off > 0; off >>= 1) {
        if (t < off) sdata[t] += sdata[t + off];
        __syncthreads();
    }
    if (t == 0) norms[b] = sdata[0];
}

__global__ void rmac_scale(float* __restrict__ out,
                           const float* __restrict__ norms, int nElem) {
    const int i = blockIdx.x * blockDim.x + threadIdx.x;
    if (i < nElem) {
        const int b = i / OUT_COLS;
        const float inv = 1.0f / fmaxf(sqrtf(norms[b]), 1e-12f);
        out[i] = out[i] * inv;
    }
}

extern "C" void kernel_launch(void* const* d_in, const int* in_sizes, int n_in,
                              void* d_out, int out_size, void* d_ws, size_t ws_size,
                              hipStream_t stream) {
    const float* x  = (const float*)d_in[0];
    float* out      = (float*)d_out;
    float* partials = (float*)d_ws;          // 8192 floats
    float* norms    = partials + WAVES;      // 64 floats

    // 8192 waves, 4 waves (128 threads) per block -> 2048 blocks
    rmac_pool_wmma<<<WAVES / 4, 128, 0, stream>>>(x, out, partials);
    rmac_norm_reduce<<<BATCH, WAVES_PER_B, 0, stream>>>(partials, norms);

    const int nElem = BATCH * OUT_COLS;
    rmac_scale<<<(nElem + 255) / 256, 256, 0, stream>>>(out, norms, nElem);
}